// SortedConv2DWithShift_27204322853079
// MI455X (gfx1250) — compile-verified
//
#include <hip/hip_runtime.h>
#include <hip/hip_bf16.h>

typedef __attribute__((ext_vector_type(2))) float v2f;
typedef __attribute__((ext_vector_type(4))) float v4f;
typedef __attribute__((ext_vector_type(8))) float v8f;

namespace {
constexpr int Bn = 32, Hn = 128, Wn = 128, Cin = 32, Fn = 64;
constexpr int BAND    = 4;          // image rows per workgroup (one pooled-block band)
constexpr int RROWS   = BAND + 2;   // staged rows incl. halo
constexpr int RCOLS   = Wn + 2;     // staged cols incl. halo
constexpr int RSTRIDE = 36;         // padded per-pixel cin stride (bank-conflict pad)
constexpr int KTOT    = 100;        // 32 center + 32 edge + 32 corner + 4 asym deltas
constexpr int NKSTEP  = KTOT / 4;   // wmma_f32_16x16x4_f32 k-steps
constexpr int MCHUNK  = 256;        // GEMM rows per chunk (2 image rows)

// LDS layout (float offsets)
constexpr int OFF_RAW  = 0;                                  // 6*130*36 = 28080 f
constexpr int OFF_S    = OFF_RAW + RROWS * RCOLS * RSTRIDE;  // channel sums, 780 f
constexpr int OFF_WK   = OFF_S + RROWS * RCOLS;              // Wk pair-interleaved [50][64][2]
constexpr int OFF_FEAT = OFF_WK + KTOT * Fn;                 // feat[256][100]
constexpr int OFF_BIAS = OFF_FEAT + MCHUNK * KTOT;           // bias[64]
constexpr int OFF_AMAP = OFF_BIAS + Fn;                      // argmax per pixel, 512 i32
constexpr int OFF_SH   = OFF_AMAP + BAND * Wn;               // shift per block, 32 i32
constexpr int LDS_FLOATS = OFF_SH + (Wn / 4);
constexpr size_t LDS_BYTES = (size_t)LDS_FLOATS * 4;         // ~246 KB < 320 KB/WGP
}  // namespace

__global__ __launch_bounds__(256, 1)
void sorted_conv_shift_fused(const float* __restrict__ in,
                             const float* __restrict__ sym_a,
                             const float* __restrict__ sym_b,
                             const float* __restrict__ sym_c,
                             const float* __restrict__ bias,
                             const float* __restrict__ scale_s,
                             const float* __restrict__ scale_a,
                             float* __restrict__ out) {
  extern __shared__ float smem[];
  float* raw   = smem + OFF_RAW;
  float* stile = smem + OFF_S;
  float* wk    = smem + OFF_WK;   // wk[((k>>1)*64 + n)*2 + (k&1)]
  float* feat  = smem + OFF_FEAT;
  float* biasl = smem + OFF_BIAS;
  int*   amap  = (int*)(smem + OFF_AMAP);
  int*   shl   = (int*)(smem + OFF_SH);

  const int tid  = threadIdx.x;
  const int b    = blockIdx.x >> 5;
  const int band = blockIdx.x & 31;
  const int h0   = band * BAND;

  const float ss = scale_s[0];
  const float sa = scale_a[0];

  // ---- Stage 1: stage input band (zero halo), build fused Wk, stage bias ----
  for (int idx = tid; idx < RROWS * RCOLS * (Cin / 4); idx += 256) {
    const int pix = idx >> 3, q = idx & 7;
    const int iy = pix / RCOLS, ix = pix - iy * RCOLS;
    const int h = h0 - 1 + iy, w = ix - 1;
    v4f v = {0.f, 0.f, 0.f, 0.f};
    if (h >= 0 && h < Hn && w >= 0 && w < Wn)
      v = *(const v4f*)(in + ((size_t)((b * Hn + h) * Wn + w)) * Cin + q * 4);
    *(v4f*)(raw + pix * RSTRIDE + q * 4) = v;
  }
  for (int idx = tid; idx < KTOT * Fn; idx += 256) {
    const int k = idx >> 6, f = idx & 63;
    float v;
    if (k < 96) {  // rows 0-31: center (sym_c), 32-63: edge (sym_b), 64-95: corner (sym_a)
      const int cin = k & 31;
      const float* sp = (k < 32) ? sym_c : (k < 64) ? sym_b : sym_a;
      v = ss * sp[cin * Fn + f];
    } else {       // rows 96-99: cin-independent asym bank a,b,c,d (pre-scaled)
      const float pi = 3.14159265358979323846f;
      const float t  = pi * (float)f / 63.0f;
      const float s8 = 2.8284712474619e0f * 0.999999f;  // placeholder avoided below
      const float ph = 9.0f * pi / 4.0f;
      const float sq8 = 2.8284271247461903f;
      float wv;
      if (k == 96)      wv = -sq8 * cosf(t - ph);
      else if (k == 97) wv = -2.0f * sinf(t);
      else if (k == 98) wv = -sq8 * sinf(t - ph);
      else              wv = -2.0f * cosf(t);
      (void)s8;
      v = sa * wv;
    }
    // pair-interleaved store: rows k and k+1 adjacent per column
    wk[(((k >> 1) * Fn + f) << 1) + (k & 1)] = v;
  }
  if (tid < Fn) biasl[tid] = bias[tid];
  __syncthreads();

  // ---- Stage 2: per-position channel sums (feed the asym-delta features) ----
  for (int idx = tid; idx < RROWS * RCOLS; idx += 256) {
    const float* p = raw + idx * RSTRIDE;
    v4f s4 = {0.f, 0.f, 0.f, 0.f};
#pragma unroll
    for (int q = 0; q < 8; ++q) s4 += *(const v4f*)(p + q * 4);
    stile[idx] = s4.x + s4.y + s4.z + s4.w;
  }
  __syncthreads();

  const int wv_  = tid >> 5;
  const int lane = tid & 31;
  const int l16  = lane & 15;
  const int koff = (lane >= 16) ? 2 : 0;  // A/B lanes 16-31 hold K+2,K+3

  v8f acc[2][2][4];
#pragma unroll
  for (int c = 0; c < 2; ++c)
#pragma unroll
    for (int i = 0; i < 2; ++i)
#pragma unroll
      for (int n = 0; n < 4; ++n) acc[c][i][n] = (v8f){0, 0, 0, 0, 0, 0, 0, 0};

#pragma unroll
  for (int ch = 0; ch < 2; ++ch) {
    // ---- Stage 3: features + x_a argmax for this 256-pixel chunk ----
    {
      const int p  = ch * MCHUNK + tid;
      const int ty = p >> 7, tx = p & 127;
      float* fr = feat + tid * KTOT;
      const float* r = raw + (ty * RCOLS + tx) * RSTRIDE;
#pragma unroll
      for (int q = 0; q < 8; ++q) {
        const v4f c0 = *(const v4f*)(r + (1 * RCOLS + 1) * RSTRIDE + q * 4);
        const v4f e  = *(const v4f*)(r + (0 * RCOLS + 1) * RSTRIDE + q * 4)
                     + *(const v4f*)(r + (2 * RCOLS + 1) * RSTRIDE + q * 4)
                     + *(const v4f*)(r + (1 * RCOLS + 0) * RSTRIDE + q * 4)
                     + *(const v4f*)(r + (1 * RCOLS + 2) * RSTRIDE + q * 4);
        const v4f k4 = *(const v4f*)(r + (0 * RCOLS + 0) * RSTRIDE + q * 4)
                     + *(const v4f*)(r + (0 * RCOLS + 2) * RSTRIDE + q * 4)
                     + *(const v4f*)(r + (2 * RCOLS + 0) * RSTRIDE + q * 4)
                     + *(const v4f*)(r + (2 * RCOLS + 2) * RSTRIDE + q * 4);
        *(v4f*)(fr + q * 4)      = c0;
        *(v4f*)(fr + 32 + q * 4) = e;
        *(v4f*)(fr + 64 + q * 4) = k4;
      }
      const float d0 = stile[(ty + 0) * RCOLS + tx + 0] - stile[(ty + 2) * RCOLS + tx + 2];
      const float d1 = stile[(ty + 0) * RCOLS + tx + 1] - stile[(ty + 2) * RCOLS + tx + 1];
      const float d2 = stile[(ty + 0) * RCOLS + tx + 2] - stile[(ty + 2) * RCOLS + tx + 0];
      const float d3 = stile[(ty + 1) * RCOLS + tx + 0] - stile[(ty + 1) * RCOLS + tx + 2];
      *(v4f*)(fr + 96) = (v4f){d0, d1, d2, d3};
      float best = -3.402823466e+38f;
      int bi = 0;
      for (int f = 0; f < Fn; ++f) {  // x_a pre-ReLU, no bias; first-max wins like argmax
        const v2f wab = *(const v2f*)(wk + ((48 * Fn + f) << 1));  // rows 96,97
        const v2f wcd = *(const v2f*)(wk + ((49 * Fn + f) << 1));  // rows 98,99
        const float xa = wab.x * d0 + wab.y * d1 + wcd.x * d2 + wcd.y * d3;
        if (xa > best) { best = xa; bi = f; }
      }
      amap[p] = bi;
    }
    __syncthreads();

    // ---- Stage 4: WMMA GEMM, D[256x64] = feat[256x100] x Wk[100x64] (fp32) ----
    for (int ks = 0; ks < NKSTEP; ++ks) {
      const int k0 = ks * 4;
      const v2f a0 = *(const v2f*)(feat + (wv_ * 32 + l16) * KTOT + k0 + koff);
      const v2f a1 = *(const v2f*)(feat + (wv_ * 32 + 16 + l16) * KTOT + k0 + koff);
      // pair-interleaved Wk: one b64 per B fragment, conflict-free half-wave banks
      const float* wp = wk + ((k0 + koff) >> 1) * (Fn * 2) + l16 * 2;
#pragma unroll
      for (int nt = 0; nt < 4; ++nt) {
        const v2f bv = *(const v2f*)(wp + nt * 32);
        acc[ch][0][nt] = __builtin_amdgcn_wmma_f32_16x16x4_f32(
            false, a0, false, bv, (short)0, acc[ch][0][nt], false, false);
        acc[ch][1][nt] = __builtin_amdgcn_wmma_f32_16x16x4_f32(
            false, a1, false, bv, (short)0, acc[ch][1][nt], false, false);
      }
    }
    __syncthreads();  // feat buffer is reused by the next chunk
  }

  // ---- Stage 5: pool argmax over geometric 4x4 blocks -> m and shift ----
  if (tid < 32) {
    int s = 0;
#pragma unroll
    for (int dy = 0; dy < 4; ++dy)
#pragma unroll
      for (int dx = 0; dx < 4; ++dx) s += amap[dy * Wn + tid * 4 + dx];
    const int m = s >> 4;  // exact trunc(mean of 16 small ints)
    shl[tid] = m & (Fn - 1);
    int* mout = (int*)out + (size_t)Bn * Hn * Wn * Fn;
    mout[b * 1024 + band * 32 + tid] = m;
  }
  __syncthreads();

  // ---- Stage 6: bias + ReLU + reference's strip-indexed channel roll ----
  // pixel (h,w) uses pooled block (h>>2, 8*(h&3) + w/16); h&3 == ty in this band
#pragma unroll
  for (int ch = 0; ch < 2; ++ch) {
#pragma unroll
    for (int i = 0; i < 2; ++i) {
#pragma unroll
      for (int v = 0; v < 8; ++v) {
        const int row = ch * MCHUNK + wv_ * 32 + i * 16 + v + ((lane >= 16) ? 8 : 0);
        const int ty = row >> 7, tx = row & 127;
        const int sh = shl[8 * ty + (tx >> 4)];
        float* op = out + ((size_t)((b * Hn + h0 + ty) * Wn + tx)) * Fn;
#pragma unroll
        for (int nt = 0; nt < 4; ++nt) {
          const int col = nt * 16 + l16;
          float val = acc[ch][i][nt][v] + biasl[col];
          val = fmaxf(val, 0.0f);
          op[(col - sh) & (Fn - 1)] = val;  // out[c] = x[(c+shift)%F]
        }
      }
    }
  }
}

extern "C" void kernel_launch(void* const* d_in, const int* in_sizes, int n_in,
                              void* d_out, int out_size, void* d_ws, size_t ws_size,
                              hipStream_t stream) {
  (void)in_sizes; (void)n_in; (void)d_ws; (void)ws_size; (void)out_size;
  static_assert(LDS_BYTES <= 320 * 1024, "LDS over WGP limit");
  hipFuncSetAttribute((const void*)sorted_conv_shift_fused,
                      hipFuncAttributeMaxDynamicSharedMemorySize, (int)LDS_BYTES);
  sorted_conv_shift_fused<<<Bn * (Hn / BAND), 256, LDS_BYTES, stream>>>(
      (const float*)d_in[0], (const float*)d_in[1], (const float*)d_in[2],
      (const float*)d_in[3], (const float*)d_in[4], (const float*)d_in[5],
      (const float*)d_in[6], (float*)d_out);
}